// Extractor_91104846283028
// MI455X (gfx1250) — compile-verified
//
#include <hip/hip_runtime.h>

#define GRID_RES 64
#define C_CH 32
#define H_IMG 512
#define W_IMG 512

typedef float v2f __attribute__((ext_vector_type(2)));
typedef float v8f __attribute__((ext_vector_type(8)));

// Bilinear sampling + output write, shared by main (WMMA) and tail (VALU) paths.
__device__ __forceinline__ void sample_point(
    int p, int N, int bi,
    float camx, float camy, float camz, float camw,
    const float* __restrict__ K,
    const float* __restrict__ feat,
    float* __restrict__ outF)
{
    // cam = cam / cam.w  (reference order)
    float icw = 1.0f / camw;
    float cx = camx * icw, cy = camy * icw, cz = camz * icw;
    // img_h = cam[:3] @ K.T
    float u  = K[0]*cx + K[1]*cy + K[2]*cz;
    float v  = K[3]*cx + K[4]*cy + K[5]*cz;
    float zz = K[6]*cx + K[7]*cy + K[8]*cz;
    float iz = 1.0f / zz;
    float ix = u * iz;   // image_x
    float iy = v * iz;   // image_y

    float i0f = floorf(iy), j0f = floorf(ix);
    float di = iy - i0f, dj = ix - j0f;

    float i0c = fminf(fmaxf(i0f,        0.0f), (float)(H_IMG - 1));
    float i1c = fminf(fmaxf(i0f + 1.0f, 0.0f), (float)(H_IMG - 1));
    float j0c = fminf(fmaxf(j0f,        0.0f), (float)(W_IMG - 1));
    float j1c = fminf(fmaxf(j0f + 1.0f, 0.0f), (float)(W_IMG - 1));
    int i0 = (int)i0c, i1 = (int)i1c, j0 = (int)j0c, j1 = (int)j1c;

    int o00 = i0 * W_IMG + j0;
    int o01 = i0 * W_IMG + j1;
    int o10 = i1 * W_IMG + j0;
    int o11 = i1 * W_IMG + j1;

    // Bilinear weights: (1-dj)*((1-di)f00 + di f10) + dj*((1-di)f01 + di f11)
    float omdi = 1.0f - di, omdj = 1.0f - dj;
    float w00 = omdi * omdj;
    float w10 = di   * omdj;
    float w01 = omdi * dj;
    float w11 = di   * dj;

    const size_t HW = (size_t)H_IMG * W_IMG;
    const float* __restrict__ Fb = feat + (size_t)bi * ((size_t)C_CH * HW);
    float* __restrict__ orow = outF + (size_t)p * C_CH;

    // 32 channels: 4 gathers + 3 FMAs each; float4 stores (row is 128B-aligned).
#pragma unroll
    for (int cg = 0; cg < C_CH / 4; ++cg) {
        float acc[4];
#pragma unroll
        for (int t = 0; t < 4; ++t) {
            const float* __restrict__ Fc = Fb + (size_t)(cg * 4 + t) * HW;
            float f00 = Fc[o00];
            float f10 = Fc[o10];
            float f01 = Fc[o01];
            float f11 = Fc[o11];
            acc[t] = fmaf(w11, f11, fmaf(w01, f01, fmaf(w10, f10, w00 * f00)));
        }
        float4 r = make_float4(acc[0], acc[1], acc[2], acc[3]);
        *(float4*)(orow + cg * 4) = r;
    }
    // depth = cam.z after division by w
    outF[(size_t)N * C_CH + (size_t)p] = cz;
}

// Main kernel: block=160 (5 full wave32s), every lane owns a valid point so
// EXEC is all-1s for WMMA. cam(4xN) = P(4x4) x pts_h(4xN) via two
// V_WMMA_F32_16X16X4_F32 per wave (16 points each).
__global__ void __launch_bounds__(160, 1) ext_main(
    const int* __restrict__ dc, const float* __restrict__ feat,
    const float* __restrict__ K, const float* __restrict__ P,
    float* __restrict__ outF, int N)
{
    int p = blockIdx.x * 160 + threadIdx.x;
    int4 q = ((const int4*)dc)[p];

    const float s = 2.0f / (float)(GRID_RES - 1);
    float xw = (float)q.y * s - 1.0f;
    float yw = (float)q.z * s - 1.0f;
    float zw = (float)q.w * s - 1.0f;

    unsigned lane = threadIdx.x & 31u;
    unsigned m    = lane & 15u;           // row (A) / column (B,D) index
    bool     hi   = lane >= 16u;          // lane half selects K pair

    // A = P padded to 16x4. Layout: VGPR0 = K{0|2}, VGPR1 = K{1|3} by half.
    float a0 = 0.0f, a1 = 0.0f;
    if (m < 4u) {
        a0 = P[m * 4u + (hi ? 2u : 0u)];
        a1 = P[m * 4u + (hi ? 3u : 1u)];
    }
    v2f A; A.x = a0; A.y = a1;

    // B for points 0..15 of the wave: column m holds lane m's homog coords.
    float sx = __shfl(xw, (int)m, 32);
    float sy = __shfl(yw, (int)m, 32);
    float sz = __shfl(zw, (int)m, 32);
    v2f Ba; Ba.x = hi ? sz : sx; Ba.y = hi ? 1.0f : sy;

    // B for points 16..31.
    float sx2 = __shfl(xw, (int)m + 16, 32);
    float sy2 = __shfl(yw, (int)m + 16, 32);
    float sz2 = __shfl(zw, (int)m + 16, 32);
    v2f Bb; Bb.x = hi ? sz2 : sx2; Bb.y = hi ? 1.0f : sy2;

    v8f Z = {};
    v8f Da = __builtin_amdgcn_wmma_f32_16x16x4_f32(
        false, A, false, Ba, (short)0, Z, false, false);
    v8f Db = __builtin_amdgcn_wmma_f32_16x16x4_f32(
        false, A, false, Bb, (short)0, Z, false, false);

    // D[j] (j<4) at lane n = cam component j of wave-point n (Da) / n+16 (Db).
    // Low lanes read their own Da element; high lanes fetch Db from lane m.
    float t0 = __shfl(Db[0], (int)m, 32);
    float t1 = __shfl(Db[1], (int)m, 32);
    float t2 = __shfl(Db[2], (int)m, 32);
    float t3 = __shfl(Db[3], (int)m, 32);
    float camx = hi ? t0 : Da[0];
    float camy = hi ? t1 : Da[1];
    float camz = hi ? t2 : Da[2];
    float camw = hi ? t3 : Da[3];

    sample_point(p, N, q.x, camx, camy, camz, camw, K, feat, outF);
}

// Tail kernel (bounds-checked, pure VALU 4x4 matvec). Only launches when
// N % 160 != 0 (never for N = 2,000,000); keeps kernel_launch general.
__global__ void __launch_bounds__(256, 1) ext_tail(
    const int* __restrict__ dc, const float* __restrict__ feat,
    const float* __restrict__ K, const float* __restrict__ P,
    float* __restrict__ outF, int N, int start)
{
    int p = start + blockIdx.x * 256 + threadIdx.x;
    if (p >= N) return;
    int4 q = ((const int4*)dc)[p];

    const float s = 2.0f / (float)(GRID_RES - 1);
    float xw = (float)q.y * s - 1.0f;
    float yw = (float)q.z * s - 1.0f;
    float zw = (float)q.w * s - 1.0f;

    float camx = P[0]  * xw + P[1]  * yw + P[2]  * zw + P[3];
    float camy = P[4]  * xw + P[5]  * yw + P[6]  * zw + P[7];
    float camz = P[8]  * xw + P[9]  * yw + P[10] * zw + P[11];
    float camw = P[12] * xw + P[13] * yw + P[14] * zw + P[15];

    sample_point(p, N, q.x, camx, camy, camz, camw, K, feat, outF);
}

extern "C" void kernel_launch(void* const* d_in, const int* in_sizes, int n_in,
                              void* d_out, int out_size, void* d_ws, size_t ws_size,
                              hipStream_t stream) {
    const int*   dc   = (const int*)d_in[0];     // dense_coords (N,4) int32
    const float* feat = (const float*)d_in[1];   // pixel_features (B,C,H,W) f32
    const float* K    = (const float*)d_in[2];   // (3,3) f32
    const float* P    = (const float*)d_in[3];   // (4,4) f32
    float* outF = (float*)d_out;                 // [N*C feature | N depth]

    int N = in_sizes[0] / 4;
    int nfull = N / 160;
    if (nfull > 0) {
        ext_main<<<nfull, 160, 0, stream>>>(dc, feat, K, P, outF, N);
    }
    int start = nfull * 160;
    int rem = N - start;
    if (rem > 0) {
        ext_tail<<<(rem + 255) / 256, 256, 0, stream>>>(dc, feat, K, P, outF, N, start);
    }
}